// decoder_gru_14929306320898
// MI455X (gfx1250) — compile-verified
//
#include <hip/hip_runtime.h>
#include <hip/hip_bf16.h>
#include <math.h>

// ---------------------------------------------------------------------------
// Problem constants (from reference): EMB=50, feature=dense=seq=4096, TF=1
// ---------------------------------------------------------------------------
#define DENSE  4096
#define FEAT   4096
#define SEQ    4096
#define EMB    50
#define G      150      // 3*EMB gates
#define GP     160      // gates padded to 16 multiple
#define KTOT   8192     // dense + feat
#define H2S    64       // padded K stride for h2 / W2 buffers

typedef __attribute__((ext_vector_type(2))) float v2f;
typedef __attribute__((ext_vector_type(8))) float v8f;

__device__ __forceinline__ float sigmoidf_(float x) {
    return 1.0f / (1.0f + __expf(-x));
}

// ---------------------------------------------------------------------------
// Kernel 0: pad weights into workspace so GEMM hot loops have NO guards.
//   Wih1p : (160, 8192)  rows 150..159 = 0
//   W2p   : (4096, 64)   cols 50..63   = 0
//   zrow  : (4096,)      zeros (stands in for fb at t==0)
// ---------------------------------------------------------------------------
__global__ __launch_bounds__(256)
void prep_pad_kernel(const float* __restrict__ W_ih1,   // (150, 8192)
                     const float* __restrict__ W2,      // (4096, 50)
                     float* __restrict__ Wih1p,
                     float* __restrict__ W2p,
                     float* __restrict__ zrow)
{
    const int i = blockIdx.x * blockDim.x + threadIdx.x;
    // 160*8192 = 1,310,720 threads total
    Wih1p[i] = (i < G * KTOT) ? W_ih1[i] : 0.f;
    if (i < FEAT * H2S) {
        const int n = i >> 6, k = i & 63;
        W2p[i] = (k < EMB) ? W2[(size_t)n * EMB + k] : 0.f;
    }
    if (i < DENSE) zrow[i] = 0.f;
}

// ---------------------------------------------------------------------------
// Kernel 0b: db[g] = b_ih1[g] + dense . W_ih1[g, 0:4096]
//   The dense half of the concatenated GRU input is identical for all 4096
//   timesteps, so its contribution is a single 160-vector computed ONCE here
//   (instead of redundantly inside every GEMM tile).  One block per gate row.
// ---------------------------------------------------------------------------
__global__ __launch_bounds__(256)
void dense_gemv_kernel(const float* __restrict__ dense,   // (4096,)
                       const float* __restrict__ W_ih1,   // (150, 8192)
                       const float* __restrict__ b_ih1,   // (150,)
                       float* __restrict__ db)            // (160,)
{
    __shared__ float red[256];
    const int g   = blockIdx.x;          // 0..159
    const int tid = threadIdx.x;

    float acc = 0.f;
    if (g < G) {
        const float* wrow = W_ih1 + (size_t)g * KTOT;    // dense part = cols 0..4095
        for (int k = tid; k < DENSE; k += 256)
            acc = fmaf(dense[k], wrow[k], acc);
    }
    red[tid] = acc;
    __syncthreads();
    #pragma unroll
    for (int s = 128; s > 0; s >>= 1) {
        if (tid < s) red[tid] += red[tid + s];
        __syncthreads();
    }
    if (tid == 0) db[g] = (g < G) ? (red[0] + b_ih1[g]) : 0.f;
}

// ---------------------------------------------------------------------------
// Kernel 1: GI1[t, g] = fb_t @ W_ih1[:, 4096:]^T + db
//   fp32 WMMA 16x16x4 over K=4096 (feedback half only; dense half folded
//   into the db epilogue bias).  One wave -> 16 rows x 32 gates (2 C tiles,
//   A pair reused).  All loads unconditional (padded operands).
// ---------------------------------------------------------------------------
__global__ __launch_bounds__(128)
void gi1_gemm_kernel(const float* __restrict__ onehot,  // (4096, 4096)
                     const float* __restrict__ zrow,    // (4096,) zeros
                     const float* __restrict__ Wih1p,   // (160, 8192) padded
                     const float* __restrict__ db,      // (160,) dense part + bias
                     float* __restrict__ GI1)           // (4096, 160)
{
    const int lane = threadIdx.x;          // 0..31
    const int half = lane >> 4;            // 0 or 1
    const int nr   = lane & 15;

    const int row0 = (blockIdx.x * 4 + threadIdx.y) * 16;   // timestep tile
    const int g0   = blockIdx.y * 32;                       // gate pair base
    const int m    = row0 + (lane & 15);                    // A row (timestep)

    // fb[t] = onehot[t-1] for t>=1, zeros at t==0
    const float* arow  = (m == 0) ? zrow : (onehot + (size_t)(m - 1) * FEAT);
    const float* brow0 = Wih1p + (size_t)(g0 + nr) * KTOT + DENSE;
    const float* brow1 = brow0 + (size_t)16 * KTOT;

    v8f c0 = {0.f, 0.f, 0.f, 0.f, 0.f, 0.f, 0.f, 0.f};
    v8f c1 = c0;

    #pragma unroll 4
    for (int k0 = 0; k0 < FEAT; k0 += 4) {
        const int ka = k0 + 2 * half;
        v2f a  = *(const v2f*)(arow + ka);
        v2f b0 = *(const v2f*)(brow0 + ka);
        v2f b1 = *(const v2f*)(brow1 + ka);
        c0 = __builtin_amdgcn_wmma_f32_16x16x4_f32(false, a, false, b0, (short)0, c0, false, false);
        c1 = __builtin_amdgcn_wmma_f32_16x16x4_f32(false, a, false, b1, (short)0, c1, false, false);
    }

    // D layout: VGPR v -> M = v + 8*half, N = lane&15
    const int ga = g0 + nr;
    const int gb = ga + 16;
    const float biasa = db[ga];
    const float biasb = db[gb];
    #pragma unroll
    for (int v = 0; v < 8; ++v) {
        const int mo = row0 + v + 8 * half;
        GI1[(size_t)mo * GP + ga] = c0[v] + biasa;
        GI1[(size_t)mo * GP + gb] = c1[v] + biasb;
    }
}

// ---------------------------------------------------------------------------
// Kernel 2: sequential GRU scan (the only truly serial part).
//   One workgroup, 512 threads.  Weight rows live in VGPRs, h1/h2 in LDS.
//   Writes h2_t into H2pad (stride 64, cols 50..63 zeroed) for kernel 3.
// ---------------------------------------------------------------------------
__global__ __launch_bounds__(512)
void gru_scan_kernel(const float* __restrict__ GI1,     // (4096, 160)
                     const float* __restrict__ W_hh1,   // (150, 50)
                     const float* __restrict__ b_hh1,   // (150,)
                     const float* __restrict__ W_ih2,   // (150, 50)
                     const float* __restrict__ b_ih2,   // (150,)
                     const float* __restrict__ W_hh2,   // (150, 50)
                     const float* __restrict__ b_hh2,   // (150,)
                     float* __restrict__ H2)            // (4096, 64) padded
{
    __shared__ float sh1[EMB];
    __shared__ float sh2[EMB];
    __shared__ float sgh1[G];
    __shared__ float sgi2[G];
    __shared__ float sgh2[G];

    const int tid = threadIdx.x;

    // Assign roles and pull this thread's weight row into registers.
    int role = -1, g = -1;
    const float* wrow = nullptr;
    float bias = 0.f;
    if (tid < G)                      { role = 0; g = tid;       wrow = W_hh1 + (size_t)g * EMB; bias = b_hh1[g]; }
    else if (tid >= 160 && tid < 310) { role = 1; g = tid - 160; wrow = W_ih2 + (size_t)g * EMB; bias = b_ih2[g]; }
    else if (tid >= 320 && tid < 470) { role = 2; g = tid - 320; wrow = W_hh2 + (size_t)g * EMB; bias = b_hh2[g]; }

    float w[EMB];
    if (role >= 0) {
        #pragma unroll
        for (int k = 0; k < EMB; ++k) w[k] = wrow[k];
    }

    if (tid < EMB) { sh1[tid] = 0.f; sh2[tid] = 0.f; }
    __syncthreads();

    for (int t = 0; t < SEQ; ++t) {
        const float* gi1 = GI1 + (size_t)t * GP;

        // Prefetch next step's precomputed input gates while we crunch this one.
        if (role == 0 && t + 1 < SEQ)
            __builtin_prefetch(GI1 + (size_t)(t + 1) * GP + g, 0, 0);

        // Phase 1: gh1 = W_hh1 @ h1 + b ; gh2 = W_hh2 @ h2 + b (independent)
        if (role == 0) {
            float acc = bias;
            #pragma unroll
            for (int k = 0; k < EMB; ++k) acc = fmaf(w[k], sh1[k], acc);
            sgh1[g] = acc;
        } else if (role == 2) {
            float acc = bias;
            #pragma unroll
            for (int k = 0; k < EMB; ++k) acc = fmaf(w[k], sh2[k], acc);
            sgh2[g] = acc;
        }
        __syncthreads();

        // h1 update (PyTorch gate order r, z, n)
        if (tid < EMB) {
            const float r = sigmoidf_(gi1[tid]           + sgh1[tid]);
            const float z = sigmoidf_(gi1[EMB + tid]     + sgh1[EMB + tid]);
            const float n = tanhf    (gi1[2 * EMB + tid] + r * sgh1[2 * EMB + tid]);
            sh1[tid] = (1.0f - z) * n + z * sh1[tid];
        }
        __syncthreads();

        // Phase 2: gi2 = W_ih2 @ h1_new + b
        if (role == 1) {
            float acc = bias;
            #pragma unroll
            for (int k = 0; k < EMB; ++k) acc = fmaf(w[k], sh1[k], acc);
            sgi2[g] = acc;
        }
        __syncthreads();

        // h2 update + emit padded row for the output GEMM
        if (tid < EMB) {
            const float r = sigmoidf_(sgi2[tid]           + sgh2[tid]);
            const float z = sigmoidf_(sgi2[EMB + tid]     + sgh2[EMB + tid]);
            const float n = tanhf    (sgi2[2 * EMB + tid] + r * sgh2[2 * EMB + tid]);
            const float h2n = (1.0f - z) * n + z * sh2[tid];
            sh2[tid] = h2n;
            H2[(size_t)t * H2S + tid] = h2n;
        } else if (tid < H2S) {
            H2[(size_t)t * H2S + tid] = 0.f;   // zero pad K=50..63
        }
        __syncthreads();
    }
}

// ---------------------------------------------------------------------------
// Kernel 3: OUT = H2pad(4096x50) @ W2^T + b2  -> (4096, 4096) fp32
//   fp32 WMMA 16x16x4; K padded 50 -> 52; both operands pre-padded, so the
//   13-iteration loop is pure load->wmma.  One wave -> 16 rows x 64 cols
//   (4 C tiles, A pair reused 4x).
// ---------------------------------------------------------------------------
__global__ __launch_bounds__(128)
void out_gemm_kernel(const float* __restrict__ H2,    // (4096, 64) padded
                     const float* __restrict__ W2p,   // (4096, 64) padded
                     const float* __restrict__ b2,    // (4096,)
                     float* __restrict__ OUT)         // (4096, 4096)
{
    const int lane = threadIdx.x;                     // 0..31
    const int half = lane >> 4;
    const int nr   = lane & 15;

    const int row0 = blockIdx.x * 16;
    const int n0   = (blockIdx.y * 4 + threadIdx.y) * 64;
    const int m    = row0 + (lane & 15);

    const float* arow = H2 + (size_t)m * H2S;
    const float* br0  = W2p + (size_t)(n0 + nr) * H2S;   // B[k][n] = W2p[n][k]
    const float* br1  = br0 + 16 * H2S;
    const float* br2  = br0 + 32 * H2S;
    const float* br3  = br0 + 48 * H2S;

    v8f c0 = {0.f, 0.f, 0.f, 0.f, 0.f, 0.f, 0.f, 0.f};
    v8f c1 = c0, c2 = c0, c3 = c0;

    #pragma unroll
    for (int k0 = 0; k0 < 52; k0 += 4) {
        const int ka = k0 + 2 * half;                    // pairs (50,51) are 0-padded
        v2f a   = *(const v2f*)(arow + ka);
        v2f b0  = *(const v2f*)(br0 + ka);
        v2f b1  = *(const v2f*)(br1 + ka);
        v2f b2v = *(const v2f*)(br2 + ka);
        v2f b3  = *(const v2f*)(br3 + ka);
        c0 = __builtin_amdgcn_wmma_f32_16x16x4_f32(false, a, false, b0,  (short)0, c0, false, false);
        c1 = __builtin_amdgcn_wmma_f32_16x16x4_f32(false, a, false, b1,  (short)0, c1, false, false);
        c2 = __builtin_amdgcn_wmma_f32_16x16x4_f32(false, a, false, b2v, (short)0, c2, false, false);
        c3 = __builtin_amdgcn_wmma_f32_16x16x4_f32(false, a, false, b3,  (short)0, c3, false, false);
    }

    #pragma unroll
    for (int v = 0; v < 8; ++v) {
        const int mo = row0 + v + 8 * half;
        float* orow = OUT + (size_t)mo * FEAT + n0 + nr;
        orow[ 0] = c0[v] + b2[n0 + nr];
        orow[16] = c1[v] + b2[n0 + nr + 16];
        orow[32] = c2[v] + b2[n0 + nr + 32];
        orow[48] = c3[v] + b2[n0 + nr + 48];
    }
}

// ---------------------------------------------------------------------------
// Launch
// ---------------------------------------------------------------------------
extern "C" void kernel_launch(void* const* d_in, const int* in_sizes, int n_in,
                              void* d_out, int out_size, void* d_ws, size_t ws_size,
                              hipStream_t stream) {
    // setup_inputs() dict order:
    const float* input_dense  = (const float*)d_in[0];   // (1,1,4096)
    const float* input_onehot = (const float*)d_in[1];   // (4096,4096)
    // d_in[2] = seq, d_in[3] = teacher_forcing (scalars); d_in[4..5] dead W1/b1
    const float* W_ih1 = (const float*)d_in[6];
    const float* W_hh1 = (const float*)d_in[7];
    const float* b_ih1 = (const float*)d_in[8];
    const float* b_hh1 = (const float*)d_in[9];
    const float* W_ih2 = (const float*)d_in[10];
    const float* W_hh2 = (const float*)d_in[11];
    const float* b_ih2 = (const float*)d_in[12];
    const float* b_hh2 = (const float*)d_in[13];
    const float* W2    = (const float*)d_in[14];
    const float* b2    = (const float*)d_in[15];
    float* out = (float*)d_out;

    // Workspace (floats):
    //   GI1 4096*160 | H2p 4096*64 | Wih1p 160*8192 | W2p 4096*64 | zrow 4096 | db 160
    float* GI1   = (float*)d_ws;
    float* H2p   = GI1   + (size_t)SEQ * GP;
    float* Wih1p = H2p   + (size_t)SEQ * H2S;
    float* W2p   = Wih1p + (size_t)GP * KTOT;
    float* zrow  = W2p   + (size_t)FEAT * H2S;
    float* db    = zrow  + (size_t)DENSE;

    // 0) Zero-pad weights so GEMM inner loops have no guards/divergence.
    prep_pad_kernel<<<(GP * KTOT + 255) / 256, 256, 0, stream>>>(
        W_ih1, W2, Wih1p, W2p, zrow);

    // 0b) Fold the timestep-invariant dense contribution + b_ih1 into db[].
    dense_gemv_kernel<<<GP, 256, 0, stream>>>(input_dense, W_ih1, b_ih1, db);

    // 1) Input-side GRU1 gates for all timesteps (WMMA GEMM over fb half only)
    {
        dim3 grid(SEQ / 64, GP / 32);   // 64 x 5; 4 waves/block, wave = 16x32 tile
        dim3 block(32, 4);
        gi1_gemm_kernel<<<grid, block, 0, stream>>>(
            input_onehot, zrow, Wih1p, db, GI1);
    }

    // 2) Sequential 50-dim GRU recurrence (single workgroup scan)
    gru_scan_kernel<<<1, 512, 0, stream>>>(
        GI1, W_hh1, b_hh1, W_ih2, b_ih2, W_hh2, b_hh2, H2p);

    // 3) Output projection GEMM (WMMA), writes the 64 MB result
    {
        dim3 grid(SEQ / 16, FEAT / 256);   // 256 x 16; 4 waves/block, wave = 16x64
        dim3 block(32, 4);
        out_gemm_kernel<<<grid, block, 0, stream>>>(H2p, W2p, b2, out);
    }
}